// StnlsNeighAttnMat_87110526697931
// MI455X (gfx1250) — compile-verified
//
#include <hip/hip_runtime.h>
#include <hip/hip_bf16.h>

// Problem constants (from reference): T=8, H=64, W=64, C=256, HEADS=8, hd=32, WS=7
#define TT 8
#define HH 64
#define WW 64
#define CC 256
#define HEADS 8
#define HDIM 32
#define WS 7
#define HW (HH * WW)                 // 4096
#define NPIX (TT * HW)               // 32768
#define NSLOT (3 * WS * WS)          // 147
#define SCALE 0.17677669529663687f   // 1/sqrt(32)

static const size_t ATTN_ELEMS = (size_t)HEADS * TT * HH * WW * NSLOT; // 38,535,168

typedef __attribute__((ext_vector_type(2))) float v2f;
typedef __attribute__((ext_vector_type(8))) float v8f;

// ---------------------------------------------------------------------------
// Kernel 1: qk[m, n] = sum_k x[m,k] * Wqk[n,k]   (M=32768, N=512, K=256)
// One wave per 16x128 output slab: 8 f32 WMMA accumulators, A fragment reused 8x.
// V_WMMA_F32_16X16X4_F32 layouts (ISA 7.12.2):
//   A 16x4 : lane%16 = M row; VGPRs = K {0,1} (lanes 0-15) / {2,3} (lanes 16-31)
//   B 4x16 : lane%16 = N col; VGPRs = K {0,1} (lanes 0-15) / {2,3} (lanes 16-31)
//   C 16x16: VGPR r -> M = r + (lane<16 ? 0 : 8), N = lane%16
// K is processed in chunks of 8 (2 WMMA steps): all 18 fragments are loaded
// into a register block first, then 16 WMMAs issue — one load clause + one
// loadcnt wait per 16 WMMAs instead of a wait per WMMA. The 8 B rows sit at
// constant byte offsets (j*16384) from a single base pointer, so addressing
// is one VGPR + immediate offsets (no per-iteration 64-bit adds).
// ---------------------------------------------------------------------------
__global__ void gemm_qk_wmma(const float* __restrict__ x,
                             const float* __restrict__ Wqk,
                             float* __restrict__ qk) {
    const int lane  = threadIdx.x & 31;
    const int mBase = blockIdx.x * 16;       // 2048 blocks in x
    const int nBase = blockIdx.y * 128;      // 4 blocks in y
    const int col   = lane & 15;
    const int kOff  = (lane >> 4) * 2;       // 0 for lanes 0-15, 2 for lanes 16-31

    const float* xbase = x   + (size_t)(mBase + col) * CC + kOff;
    const float* wbase = Wqk + (size_t)(nBase + col) * CC + kOff;  // + j*4096 floats per tile

    v8f acc[8] = {};

#pragma unroll 2
    for (int k0 = 0; k0 < CC; k0 += 8) {
        // ---- load phase: 2 A fragments + 16 B fragments (one clause) ----
        v2f a0 = *(const v2f*)(xbase + k0);
        v2f a1 = *(const v2f*)(xbase + k0 + 4);
        v2f b0[8], b1[8];
#pragma unroll
        for (int j = 0; j < 8; ++j) {
            b0[j] = *(const v2f*)(wbase + j * 4096 + k0);
            b1[j] = *(const v2f*)(wbase + j * 4096 + k0 + 4);
        }
        // ---- compute phase: 16 WMMAs ----
#pragma unroll
        for (int j = 0; j < 8; ++j)
            acc[j] = __builtin_amdgcn_wmma_f32_16x16x4_f32(
                false, a0, false, b0[j], (short)0, acc[j], false, false);
#pragma unroll
        for (int j = 0; j < 8; ++j)
            acc[j] = __builtin_amdgcn_wmma_f32_16x16x4_f32(
                false, a1, false, b1[j], (short)0, acc[j], false, false);
    }

    const int mOff = (lane >> 4) * 8;
#pragma unroll
    for (int j = 0; j < 8; ++j) {
        const int n = nBase + j * 16 + col;
#pragma unroll
        for (int r = 0; r < 8; ++r) {
            const int m = mBase + mOff + r;
            qk[(size_t)m * 512 + n] = acc[j][r];
        }
    }
}

// reflect matching jnp.mod (floored) + fold
__device__ __forceinline__ int reflect_idx(int idx, int L) {
    const int period = 2 * (L - 1);
    int m = idx % period;
    if (m < 0) m += period;
    return (m > L - 1) ? (period - m) : m;
}

// ---------------------------------------------------------------------------
// Kernel 2: one wave per pixel. Lanes parallel over the 147 window slots.
// qk workspace layout per pixel row (512 f32): [0..255]=q(head,f), [256..511]=k(head,f)
// Outputs (concatenated flat, f32):
//   attn   [HEADS][T][H][W][147]          at offset 0
//   flows_k[HEADS][T][H][W][147][3]       at offset ATTN_ELEMS (head-broadcast)
// Store volume (~617 MB) is the roofline floor (~27 us @ 23.3 TB/s); stores are
// non-temporal so the 64 MB qk workspace + 33 MB k-slice stay L2-resident for
// the window gathers.
// ---------------------------------------------------------------------------
__global__ void neigh_attn(const float* __restrict__ qk,
                           const float* __restrict__ flows,
                           float* __restrict__ out) {
    __shared__ float lds_q[8 * 256];    // 8 waves per block, 256 q floats each

    const int lane = threadIdx.x & 31;
    const int wave = threadIdx.x >> 5;
    const int pix  = blockIdx.x * 8 + wave;        // < 32768
    const int t    = pix >> 12;                    // / 4096
    const int hw   = pix & (HW - 1);
    const int h    = hw >> 6;
    const int w    = hw & 63;

    // Stage q for this pixel into LDS (all 8 heads, 256 floats).
    const float* qrow = qk + (size_t)pix * 512;
#pragma unroll
    for (int i = 0; i < 8; ++i)
        lds_q[wave * 256 + lane + 32 * i] = qrow[lane + 32 * i];
    __syncthreads();

    // Flow displacements for this pixel (round half-to-even == jnp.round).
    const size_t fb = (size_t)(t * 4) * HW + hw;   // flows[0,t,p,c,h,w], stride HW per c
    const int di1 = (int)rintf(flows[fb + 0 * HW]);
    const int dj1 = (int)rintf(flows[fb + 1 * HW]);
    const int di2 = (int)rintf(flows[fb + 2 * HW]);
    const int dj2 = (int)rintf(flows[fb + 3 * HW]);

    const int tp1 = (t + 1 < TT) ? t + 1 : t - 2;
    const int tp2 = (t - 1 >= 0) ? t - 1 : t + 2;

    for (int s = lane; s < NSLOT; s += 32) {
        const int slot = s / 49;
        const int r    = s - slot * 49;
        const int a    = r / 7;
        const int b    = r - a * 7;

        int tp, dtv, di, dj;
        if (slot == 0)      { tp = t;   dtv = 0;        di = 0;   dj = 0;   }
        else if (slot == 1) { tp = tp1; dtv = tp1 - t;  di = di1; dj = dj1; }
        else                { tp = tp2; dtv = tp2 - t;  di = di2; dj = dj2; }

        const int li = reflect_idx(h + di + a - 3, HH);
        const int lj = reflect_idx(w + dj + b - 3, WW);

        const float* kbase = qk + ((size_t)(tp * HW + li * 64 + lj)) * 512 + 256;

        // 8 heads: dot(q[d], k[d]) over hd=32 using float4 lanes.
#pragma unroll
        for (int d = 0; d < 8; ++d) {
            const float4* kp = reinterpret_cast<const float4*>(kbase + d * 32);
            const float4* qp = reinterpret_cast<const float4*>(&lds_q[wave * 256 + d * 32]);
            float acc = 0.f;
#pragma unroll
            for (int u = 0; u < 8; ++u) {
                float4 kv = kp[u];
                float4 qv = qp[u];
                acc = fmaf(kv.x, qv.x, acc);
                acc = fmaf(kv.y, qv.y, acc);
                acc = fmaf(kv.z, qv.z, acc);
                acc = fmaf(kv.w, qv.w, acc);
            }
            const size_t ai = ((size_t)(d * TT + t) * HW + hw) * NSLOT + s;
            __builtin_nontemporal_store(acc * SCALE, &out[ai]);
        }

        // Offsets (identical across heads; stored as f32).
        const float fv0 = (float)dtv;
        const float fv1 = (float)(li - h);
        const float fv2 = (float)(lj - w);
#pragma unroll
        for (int d = 0; d < 8; ++d) {
            const size_t fi = ATTN_ELEMS +
                (((size_t)(d * TT + t) * HW + hw) * NSLOT + s) * 3;
            __builtin_nontemporal_store(fv0, &out[fi + 0]);
            __builtin_nontemporal_store(fv1, &out[fi + 1]);
            __builtin_nontemporal_store(fv2, &out[fi + 2]);
        }
    }
}

extern "C" void kernel_launch(void* const* d_in, const int* in_sizes, int n_in,
                              void* d_out, int out_size, void* d_ws, size_t ws_size,
                              hipStream_t stream) {
    const float* x     = (const float*)d_in[0];   // (8,64,64,256)
    const float* flows = (const float*)d_in[1];   // (1,8,2,2,64,64)
    const float* Wqk   = (const float*)d_in[2];   // (512,256)
    float* out = (float*)d_out;
    float* qk  = (float*)d_ws;                    // 32768 x 512 f32 = 64 MB scratch

    // Stage 1: QK projection GEMM via f32 WMMA. 2048 x 4 waves, 16x128 slab each.
    dim3 g1(NPIX / 16, 4);
    gemm_qk_wmma<<<g1, 32, 0, stream>>>(x, Wqk, qk);

    // Stage 2: neighborhood attention scores + offsets. 1 wave per pixel.
    neigh_attn<<<NPIX / 8, 256, 0, stream>>>(qk, flows, out);
}